// NonLinearConv2d_27144193310941
// MI455X (gfx1250) — compile-verified
//
#include <hip/hip_runtime.h>
#include <stdint.h>

// NonLinearConv2d: per-pair softplus^2 difference, summed over 288 taps.
// Trans-bound VALU kernel; WMMA inapplicable (nonlinear in v-t).
// CDNA5 paths: global_load_async_to_lds_b32 + s_wait_asynccnt, global_prefetch.
// Trick: exp((v-t)/D) = [2^(vs-86)] * [2^(43-ts)] * 2^43 -> exps precomputed in
// the LDS tiles, leaving only 2 v_log_f32 transcendentals per pair.

#define NB  16
#define NCI 32
#define NH  32
#define NW  32
#define NOC 64
#define CKK 288

#define SCALE_F 19.23593387851951f                 // log2(e)/0.075
#define V0_F    1.2924697071141057e-26f            // 2^-86  (padding: v=0)
#define EH_F    5.184705528587072e21f              // exp(+50) high rail
#define TWO43_F 8796093022208.0f                   // 2^43 (exact)
#define K43_F   ((float)(8796093022208.0 * 0.2635971381157267)) // 2^43 * exp(-4/3)
#define FINAL_F 2.7025482032898204e-4f             // ALPHA * ln2^2

__device__ __forceinline__ void eval_pair(float V, float T, float& acc) {
  float p  = V * T;                                // 2^(d-43), d=(v-t)*SCALE
  float e1 = fminf(p * TWO43_F, EH_F);             // exp(min(arg,50)); underflow->0 ok
  float l1 = __builtin_amdgcn_logf(1.0f + e1);     // log2(1+e) = softplus/ln2
  acc = fmaf(l1, l1, acc);
  float e2 = fminf(p * K43_F, EH_F);               // exp(min(arg-4/3,50))
  float l2 = __builtin_amdgcn_logf(1.0f + e2);
  acc = fmaf(l2, -l2, acc);                        // neg src modifier is free
}

__global__ __launch_bounds__(256, 1)
void nlconv_kernel(const float* __restrict__ x,
                   const float* __restrict__ theta,
                   float* __restrict__ out) {
  extern __shared__ char smem[];
  float* ts_l = (float*)smem;              // [288][64]  T' = 2^(43 - clip(th)*SCALE)
  float* vs_l = ts_l + CKK * NOC;          // [3][32][34] V' = 2^(clip(x)*SCALE - 86)
  float* stg  = vs_l + 3 * NCI * 34;       // [3][1024]  raw async staging

  const int tid = threadIdx.x;
  const int b   = blockIdx.x >> 5;
  const int y   = blockIdx.x & 31;

  // Warm theta into L2 (global_prefetch_b8); 72KB = 576 x 128B lines
  for (int i = tid; i < 576; i += 256)
    __builtin_prefetch((const char*)theta + (size_t)i * 128, 0, 3);

  // Async copy the 3 input rows (global -> LDS staging), ASYNCcnt-tracked
  const float* xb = x + (size_t)b * (NCI * NH * NW);
  for (int r = 0; r < 3; ++r) {
    int yy = y - 1 + r;
    if ((unsigned)yy < 32u) {
      for (int i = tid; i < NCI * NW; i += 256) {          // 4 full-EXEC iterations
        int voff = (((i >> 5) * (NH * NW)) + yy * NW + (i & 31)) * 4;
        unsigned lds = (unsigned)(uintptr_t)&stg[r * 1024 + i];
        asm volatile("global_load_async_to_lds_b32 %0, %1, %2"
                     :: "v"(lds), "v"(voff), "s"(xb) : "memory");
      }
    }
  }

  // Fill tile with V'(v=0) = 2^-86 (covers W borders and out-of-range rows)
  for (int i = tid; i < 3 * NCI * 34; i += 256) vs_l[i] = V0_F;

  // theta -> LDS transposed [k][oc]: T' = 2^(43 - clip(th,1,8)*SCALE)
  for (int i = tid; i < CKK * NOC; i += 256) {
    int oc = i & 63, k = i >> 6;
    float t = __builtin_amdgcn_fmed3f(theta[oc * CKK + k], 1.0f, 8.0f);
    ts_l[i] = __builtin_amdgcn_exp2f(fmaf(t, -SCALE_F, 43.0f));
  }

  asm volatile("s_wait_asynccnt 0x0" ::: "memory");
  __syncthreads();

  // Transform staging -> padded tile: V' = 2^(clip(x,0,9)*SCALE - 86)
  for (int r = 0; r < 3; ++r) {
    int yy = y - 1 + r;
    if ((unsigned)yy < 32u) {
      for (int i = tid; i < NCI * NW; i += 256) {
        float v = __builtin_amdgcn_fmed3f(stg[r * 1024 + i], 0.0f, 9.0f);
        vs_l[(r * NCI + (i >> 5)) * 34 + (i & 31) + 1] =
            __builtin_amdgcn_exp2f(fmaf(v, SCALE_F, -86.0f));
      }
    }
  }
  __syncthreads();

  // Thread -> 4 oc (contiguous, one b128 theta load) x 2 x positions
  const int ocg = tid & 15;   // oc = 4*ocg .. 4*ocg+3
  const int xg  = tid >> 4;   // x  = xg, xg+16

  float acc[2][4] = {{0.f, 0.f, 0.f, 0.f}, {0.f, 0.f, 0.f, 0.f}};

  for (int cin = 0; cin < NCI; ++cin) {
#pragma unroll
    for (int kh = 0; kh < 3; ++kh) {
      const float* vrow = vs_l + (kh * NCI + cin) * 34;
      const float* trow = ts_l + (cin * 9 + kh * 3) * NOC + ocg * 4;
#pragma unroll
      for (int kw = 0; kw < 3; ++kw) {
        float v0 = vrow[xg + kw];            // padded index (x+kw-1)+1
        float v1 = vrow[xg + 16 + kw];
        float4 t4 = *(const float4*)(trow + kw * NOC);   // ds_load_b128
        eval_pair(v0, t4.x, acc[0][0]);
        eval_pair(v0, t4.y, acc[0][1]);
        eval_pair(v0, t4.z, acc[0][2]);
        eval_pair(v0, t4.w, acc[0][3]);
        eval_pair(v1, t4.x, acc[1][0]);
        eval_pair(v1, t4.y, acc[1][1]);
        eval_pair(v1, t4.z, acc[1][2]);
        eval_pair(v1, t4.w, acc[1][3]);
      }
    }
  }

  // out[b][oc][y][x]
  float* ob = out + ((size_t)b * NOC * NH + y) * NW;
#pragma unroll
  for (int j = 0; j < 4; ++j) {
    int oc = ocg * 4 + j;
    ob[(size_t)oc * NH * NW + xg]      = acc[0][j] * FINAL_F;
    ob[(size_t)oc * NH * NW + xg + 16] = acc[1][j] * FINAL_F;
  }
}

extern "C" void kernel_launch(void* const* d_in, const int* in_sizes, int n_in,
                              void* d_out, int out_size, void* d_ws, size_t ws_size,
                              hipStream_t stream) {
  const float* x     = (const float*)d_in[0];
  const float* theta = (const float*)d_in[1];
  float* out = (float*)d_out;
  size_t shmem = (size_t)(CKK * NOC + 3 * NCI * 34 + 3 * 1024) * sizeof(float); // 99072 B
  nlconv_kernel<<<dim3(NB * NH), dim3(256), shmem, stream>>>(x, theta, out);
}